// GraphSAGE_61649960567470
// MI455X (gfx1250) — compile-verified
//
#include <hip/hip_runtime.h>

typedef __attribute__((ext_vector_type(2))) float v2f;
typedef __attribute__((ext_vector_type(8))) float v8f;

#define NODES 100000
#define EDGES 1000000
#define FDIM  64   // IN == HID == 64

// ---------------------------------------------------------------------------
// zero workspace (float4 stores)
// ---------------------------------------------------------------------------
__global__ void zero4_kernel(float4* __restrict__ p, int n4) {
    int i = blockIdx.x * blockDim.x + threadIdx.x;
    if (i < n4) p[i] = make_float4(0.f, 0.f, 0.f, 0.f);
}

// ---------------------------------------------------------------------------
// in-degree: one thread per edge -> global_atomic_add_f32 (L2 atomic unit)
// ---------------------------------------------------------------------------
__global__ void degree_kernel(const int* __restrict__ dst, float* __restrict__ deg, int e) {
    int i = blockIdx.x * blockDim.x + threadIdx.x;
    if (i < e) atomicAdd(&deg[dst[i]], 1.0f);
}

// ---------------------------------------------------------------------------
// scatter-add aggregation: 16 threads/edge, float4 gather + 4 f32 atomics.
// x is 25.6 MB -> L2-resident on the 192 MB L2, so this is L2-bandwidth work.
// ---------------------------------------------------------------------------
__global__ void scatter_kernel(const float* __restrict__ x,
                               const int* __restrict__ src,
                               const int* __restrict__ dst,
                               float* __restrict__ agg, int e) {
    int t = blockIdx.x * blockDim.x + threadIdx.x;
    int ed = t >> 4;
    if (ed >= e) return;
    int c = (t & 15) << 2;                 // feature chunk of 4 floats
    int s = src[ed];
    int d = dst[ed];
    float4 v = *reinterpret_cast<const float4*>(x + (size_t)s * FDIM + c);
    float* ap = agg + (size_t)d * FDIM + c;
    atomicAdd(ap + 0, v.x);
    atomicAdd(ap + 1, v.y);
    atomicAdd(ap + 2, v.z);
    atomicAdd(ap + 3, v.w);
}

// ---------------------------------------------------------------------------
// Fused SAGE layer GEMM using V_WMMA_F32_16X16X4_F32 (fp32 WMMA, exact):
//   out[m,n] = relu?( (agg[m,:]/max(deg,1)) @ Wl + xin[m,:] @ Wr + b[n] )
// One wave32 per 16-row node tile. K=64 as 16 k-steps of 4. Both A operand
// sets preloaded into 64 VGPRs and reused across all N-column tiles.
// A frag (16x4 f32, ISA 7.12.2): lane l holds A[l&15][k0 + 2*(l>>4) + {0,1}].
// B frag (4x16 f32): lane l holds W[k0 + 2*(l>>4) + {0,1}][n0 + (l&15)].
// C/D (16x16 f32): lane l, VGPR j -> row j + 8*(l>>4), col n0 + (l&15).
// ---------------------------------------------------------------------------
template <int NOUT, bool RELU>
__global__ void sage_gemm_kernel(const float* __restrict__ agg,
                                 const float* __restrict__ xin,
                                 const float* __restrict__ deg,
                                 const float* __restrict__ Wl,
                                 const float* __restrict__ Wr,
                                 const float* __restrict__ bias,
                                 float* __restrict__ out, int n) {
    const int lane = threadIdx.x & 31;
    const int wave = threadIdx.x >> 5;
    const int tile = blockIdx.x * (blockDim.x >> 5) + wave;
    if (tile * 16 >= n) return;            // wave-uniform guard (N % 16 == 0)

    const int row0 = tile * 16;
    const int m    = lane & 15;            // A row / B,C column sub-index
    const int kb   = (lane >> 4) * 2;      // 0 or 2 within the K=4 slab

    const float rdeg = 1.0f / fmaxf(deg[row0 + m], 1.0f);

    // Preload A fragments for all 16 k-steps: agg (pre-scaled by 1/deg) and x.
    v2f aF[16], xF[16];
    const float* ap = agg + (size_t)(row0 + m) * FDIM + kb;
    const float* xp = xin + (size_t)(row0 + m) * FDIM + kb;
#pragma unroll
    for (int ks = 0; ks < 16; ++ks) {
        aF[ks].x = ap[ks * 4 + 0] * rdeg;
        aF[ks].y = ap[ks * 4 + 1] * rdeg;
        xF[ks].x = xp[ks * 4 + 0];
        xF[ks].y = xp[ks * 4 + 1];
    }

    constexpr int NT = (NOUT + 15) / 16;
#pragma unroll
    for (int nt = 0; nt < NT; ++nt) {
        const int n0   = nt * 16;
        const int col  = n0 + m;
        const bool cok  = (col < NOUT);
        const int  ccol = cok ? col : (NOUT - 1);     // lane select, no branch
        const float msk = cok ? 1.0f : 0.0f;          // -> EXEC stays all-1s

        const float bv = bias[ccol] * msk;
        v8f acc;
#pragma unroll
        for (int j = 0; j < 8; ++j) acc[j] = bv;

#pragma unroll
        for (int ks = 0; ks < 16; ++ks) {
            const int k = ks * 4 + kb;                // this lane's K index
            v2f bl, br;
            bl.x = Wl[(k + 0) * NOUT + ccol] * msk;
            bl.y = Wl[(k + 1) * NOUT + ccol] * msk;
            acc = __builtin_amdgcn_wmma_f32_16x16x4_f32(
                false, aF[ks], false, bl, (short)0, acc, false, false);
            br.x = Wr[(k + 0) * NOUT + ccol] * msk;
            br.y = Wr[(k + 1) * NOUT + ccol] * msk;
            acc = __builtin_amdgcn_wmma_f32_16x16x4_f32(
                false, xF[ks], false, br, (short)0, acc, false, false);
        }

        if (cok) {
#pragma unroll
            for (int j = 0; j < 8; ++j) {
                const int mr = j + 8 * (lane >> 4);
                float v = acc[j];
                if (RELU) v = fmaxf(v, 0.0f);
                out[(size_t)(row0 + mr) * NOUT + col] = v;
            }
        }
    }
}

// ---------------------------------------------------------------------------
extern "C" void kernel_launch(void* const* d_in, const int* in_sizes, int n_in,
                              void* d_out, int out_size, void* d_ws, size_t ws_size,
                              hipStream_t stream) {
    const float* x   = (const float*)d_in[0];
    const int*   ei  = (const int*)d_in[1];
    const float* W1l = (const float*)d_in[2];
    const float* W1r = (const float*)d_in[3];
    const float* b1  = (const float*)d_in[4];
    const float* W2l = (const float*)d_in[5];
    const float* W2r = (const float*)d_in[6];
    const float* b2  = (const float*)d_in[7];
    float*       out = (float*)d_out;

    const int* src = ei;
    const int* dst = ei + EDGES;

    // workspace layout: deg[N] | agg[N*64] | h[N*64]  (~52 MB)
    float* deg = (float*)d_ws;
    float* agg = deg + NODES;
    float* h   = agg + (size_t)NODES * FDIM;

    const int threads = 256;

    // --- zero deg + agg (contiguous N*(1+64) floats, divisible by 4) ---
    {
        int n4 = NODES * (1 + FDIM) / 4;
        zero4_kernel<<<(n4 + threads - 1) / threads, threads, 0, stream>>>(
            (float4*)deg, n4);
    }

    // --- in-degree ---
    degree_kernel<<<(EDGES + threads - 1) / threads, threads, 0, stream>>>(
        dst, deg, EDGES);

    // --- layer 1 aggregate + GEMM(+ReLU) ---
    {
        long long tt = (long long)EDGES * 16;
        scatter_kernel<<<(int)((tt + threads - 1) / threads), threads, 0, stream>>>(
            x, src, dst, agg, EDGES);

        int tiles  = NODES / 16;                // 6250, exact
        int wpb    = threads / 32;              // 8 waves per block
        int blocks = (tiles + wpb - 1) / wpb;
        sage_gemm_kernel<FDIM, true><<<blocks, threads, 0, stream>>>(
            agg, x, deg, W1l, W1r, b1, h, NODES);
    }

    // --- re-zero agg ---
    {
        int n4 = NODES * FDIM / 4;
        zero4_kernel<<<(n4 + threads - 1) / threads, threads, 0, stream>>>(
            (float4*)agg, n4);
    }

    // --- layer 2 aggregate + GEMM ---
    {
        long long tt = (long long)EDGES * 16;
        scatter_kernel<<<(int)((tt + threads - 1) / threads), threads, 0, stream>>>(
            h, src, dst, agg, EDGES);

        int tiles  = NODES / 16;
        int wpb    = threads / 32;
        int blocks = (tiles + wpb - 1) / wpb;
        sage_gemm_kernel<40, false><<<blocks, threads, 0, stream>>>(
            agg, h, deg, W2l, W2r, b2, out, NODES);
    }
}